// KNNModel_56581899157602
// MI455X (gfx1250) — compile-verified
//
#include <hip/hip_runtime.h>
#include <math.h>

// ---------------------------------------------------------------------------
// Fused kNN (k=3) OOD score for MI455X / gfx1250 (wave32, WMMA fp32).
//   embeddings: [4, 768, 32, 32] f32   (queries, D-strided by 1024 in memory)
//   database  : [20000, 768] f32
//   out       : [4, 1, 512, 512] f32
// Main GEMM: 128Mx128N block tile, 8 waves, each wave 32Mx64N ->
//   per K-step: 2 A-frag + 4 B-frag ds loads feed 8 v_wmma_f32_16x16x4_f32.
// ---------------------------------------------------------------------------

typedef __attribute__((ext_vector_type(2))) float v2f;
typedef __attribute__((ext_vector_type(8))) float v8f;

#define BATCH     4
#define DDIM      768
#define HW        1024          // 32*32 queries per batch image
#define NQ        4096          // total queries
#define NDB       20000
#define KNN       3
#define OH        512
#define OW        512

#define TM        128           // block M tile
#define TN        128           // block N tile
#define TK        64            // K slice
#define LDS       68            // padded LDS row stride (floats)
#define NT_TOTAL  157           // ceil(20000/128)
#define NCHUNKS   8
#define TPC       20            // tiles per chunk (last chunk: 17)

#define BIGF      3.4e38f

__device__ __forceinline__ void ins3(float d, float& t0, float& t1, float& t2) {
    if (d < t2) {
        if (d < t1) {
            t2 = t1;
            if (d < t0) { t1 = t0; t0 = d; } else { t1 = d; }
        } else {
            t2 = d;
        }
    }
}

// ------------------------- database row norms ------------------------------
__global__ void knn_x2_kernel(const float* __restrict__ db, float* __restrict__ x2) {
    __shared__ float red[256];
    const int grp = threadIdx.x >> 6;           // 4 rows per 256-thread block
    const int t   = threadIdx.x & 63;
    const int row = blockIdx.x * 4 + grp;
    float s = 0.0f;
    if (row < NDB) {
        const float* p = db + (size_t)row * DDIM;
        for (int d = t; d < DDIM; d += 64) { float v = p[d]; s += v * v; }
    }
    red[threadIdx.x] = s;
    __syncthreads();
    for (int off = 32; off > 0; off >>= 1) {
        if (t < off) red[threadIdx.x] += red[threadIdx.x + off];
        __syncthreads();
    }
    if (t == 0 && row < NDB) x2[row] = red[grp << 6];
}

// ------------------------- query norms -------------------------------------
__global__ void knn_q2_kernel(const float* __restrict__ emb, float* __restrict__ q2) {
    const int g = blockIdx.x * blockDim.x + threadIdx.x;
    if (g >= NQ) return;
    const int b = g >> 10;
    const int p = g & 1023;
    float s = 0.0f;
    for (int d = 0; d < DDIM; ++d) {
        float v = emb[((size_t)(b * DDIM + d)) * HW + p];   // coalesced across g
        s += v * v;
    }
    q2[g] = s;
}

// ------------------------- main fused GEMM + top-3 -------------------------
// grid: (32 query tiles, 8 N-chunks), block: 256 threads (8 waves, 4Mx2N grid)
__global__ void __launch_bounds__(256)
knn_main_kernel(const float* __restrict__ emb, const float* __restrict__ db,
                const float* __restrict__ q2g, const float* __restrict__ x2g,
                float* __restrict__ part) {
    __shared__ float smem[2 * TM * LDS];        // 69632 B: A slice + B slice
    float* sA = smem;                           // [m=128][k=64] padded
    float* sB = smem + TM * LDS;                // [n=128][k=64] padded

    const int tid   = threadIdx.x;
    const int lane  = tid & 31;
    const int wave  = tid >> 5;                 // 0..7
    const int wm    = wave >> 1;                // 0..3 -> 32-row M strip
    const int wn    = wave & 1;                 // 0..1 -> 64-col N strip
    const int mt    = blockIdx.x;               // query tile 0..31
    const int chunk = blockIdx.y;               // 0..7

    const int qBase = mt * TM;                  // 128-query tile (within one batch)
    const int bIdx  = qBase >> 10;
    const int pBase = qBase & 1023;

    const int tStart  = TPC * chunk;
    int nTilesC = NT_TOTAL - tStart; if (nTilesC > TPC) nTilesC = TPC;
    const int nStart  = tStart * TN;

    const int c16  = lane & 15;
    const int rHi  = (lane >> 4) * 8;           // +8 rows for upper half-wave
    const int kOff = (lane >> 4) * 2;           // K pair select

    // per-lane query norms for its 16 row slots (two 16-row sub-strips)
    float q2v[16];
#pragma unroll
    for (int i = 0; i < 2; ++i)
#pragma unroll
        for (int r = 0; r < 8; ++r)
            q2v[i * 8 + r] = q2g[qBase + wm * 32 + i * 16 + rHi + r];

    // per-lane running top-3 per row slot
    float t0[16], t1[16], t2[16];
#pragma unroll
    for (int s = 0; s < 16; ++s) { t0[s] = BIGF; t1[s] = BIGF; t2[s] = BIGF; }

    for (int nt = 0; nt < nTilesC; ++nt) {
        const int nBase = nStart + nt * TN;

        v8f acc[2][4];
#pragma unroll
        for (int i = 0; i < 2; ++i)
#pragma unroll
            for (int j = 0; j < 4; ++j) { v8f z = {}; acc[i][j] = z; }

        for (int kb = 0; kb < DDIM / TK; ++kb) {
            const int kBase = kb * TK;
            __syncthreads();

            // ---- stage A slice: emb[(b*768+k)*1024 + p], transpose into [m][k]
            {
                const int p4 = tid & 31;        // float4 index along p (128 rows)
                const int k0 = tid >> 5;        // 0..7
#pragma unroll
                for (int kk8 = 0; kk8 < 8; ++kk8) {
                    const int kk = k0 + kk8 * 8;
                    const float4 v = *(const float4*)(
                        emb + ((size_t)(bIdx * DDIM + kBase + kk)) * HW + pBase + p4 * 4);
                    sA[(p4 * 4 + 0) * LDS + kk] = v.x;
                    sA[(p4 * 4 + 1) * LDS + kk] = v.y;
                    sA[(p4 * 4 + 2) * LDS + kk] = v.z;
                    sA[(p4 * 4 + 3) * LDS + kk] = v.w;
                }
            }
            // ---- stage B slice: db rows are K-contiguous -> [n][k]
#pragma unroll
            for (int i = 0; i < 8; ++i) {
                const int idx = tid + 256 * i;          // 0..2047 float4 slots
                const int n   = idx >> 4;               // 0..127
                const int k4  = idx & 15;
                const int dbRow = nBase + n;
                float4 v = make_float4(0.f, 0.f, 0.f, 0.f);
                if (dbRow < NDB)
                    v = *(const float4*)(db + (size_t)dbRow * DDIM + kBase + k4 * 4);
                *(float4*)(&sB[n * LDS + k4 * 4]) = v;  // 272B rows: 16B aligned
            }
            __syncthreads();

            // ---- WMMA f32 16x16x4 over this K slice (EXEC all-ones here)
            const int m0 = wm * 32 + c16;
#pragma unroll
            for (int ks = 0; ks < 16; ++ks) {
                const int kk = ks * 4 + kOff;
                const v2f a0 = *(const v2f*)(&sA[m0 * LDS + kk]);
                const v2f a1 = *(const v2f*)(&sA[(m0 + 16) * LDS + kk]);
#pragma unroll
                for (int j = 0; j < 4; ++j) {
                    const v2f bf = *(const v2f*)(&sB[(wn * 64 + j * 16 + c16) * LDS + kk]);
                    acc[0][j] = __builtin_amdgcn_wmma_f32_16x16x4_f32(
                        false, a0, false, bf, (short)0, acc[0][j], false, false);
                    acc[1][j] = __builtin_amdgcn_wmma_f32_16x16x4_f32(
                        false, a1, false, bf, (short)0, acc[1][j], false, false);
                }
            }
        }

        // ---- epilogue: d2 = q2 + x2 - 2*dot ; update top-3
#pragma unroll
        for (int j = 0; j < 4; ++j) {
            const int col = nBase + wn * 64 + j * 16 + c16;
            const bool valid = (col < NDB);
            const float xc = valid ? x2g[col] : 0.0f;
#pragma unroll
            for (int i = 0; i < 2; ++i)
#pragma unroll
                for (int r = 0; r < 8; ++r) {
                    float d2 = fmaf(-2.0f, acc[i][j][r], q2v[i * 8 + r] + xc);
                    d2 = valid ? d2 : BIGF;
                    ins3(d2, t0[i * 8 + r], t1[i * 8 + r], t2[i * 8 + r]);
                }
        }
    }

    // ---- cross-lane/cross-wave merge via LDS:
    // per row: 2 N-waves x 16 column-lanes x 3 candidates = 96
    __syncthreads();
#pragma unroll
    for (int i = 0; i < 2; ++i)
#pragma unroll
        for (int r = 0; r < 8; ++r) {
            const int row  = wm * 32 + i * 16 + rHi + r;
            const int base = row * 96 + (wn * 16 + c16) * 3;
            smem[base + 0] = t0[i * 8 + r];
            smem[base + 1] = t1[i * 8 + r];
            smem[base + 2] = t2[i * 8 + r];
        }
    __syncthreads();

    if (tid < TM) {
        float a0 = BIGF, a1 = BIGF, a2 = BIGF;
        const float* c = &smem[tid * 96];
        for (int s = 0; s < 96; ++s) ins3(c[s], a0, a1, a2);
        const int gq = qBase + tid;
        float* o = part + ((size_t)chunk * NQ + gq) * KNN;
        o[0] = a0; o[1] = a1; o[2] = a2;
    }
}

// ------------------------- merge chunks -> ood map -------------------------
__global__ void knn_merge_kernel(const float* __restrict__ part, float* __restrict__ ood) {
    const int q = blockIdx.x * blockDim.x + threadIdx.x;
    if (q >= NQ) return;
    float a0 = BIGF, a1 = BIGF, a2 = BIGF;
    for (int c = 0; c < NCHUNKS; ++c) {
        const float* p = part + ((size_t)c * NQ + q) * KNN;
        ins3(p[0], a0, a1, a2);
        ins3(p[1], a0, a1, a2);
        ins3(p[2], a0, a1, a2);
    }
    const float d0 = sqrtf(fmaxf(a0, 1e-12f));
    const float d1 = sqrtf(fmaxf(a1, 1e-12f));
    const float d2 = sqrtf(fmaxf(a2, 1e-12f));
    ood[q] = (d0 + d1 + d2) * (1.0f / 3.0f);
}

// ------------------------- half-pixel bilinear 32x32 -> 512x512 ------------
__global__ void knn_upsample_kernel(const float* __restrict__ ood, float* __restrict__ out) {
    const int idx = blockIdx.x * blockDim.x + threadIdx.x;
    if (idx >= BATCH * OH * OW) return;
    const int b  = idx / (OH * OW);
    const int oy = (idx / OW) % OH;
    const int ox = idx % OW;

    const float scale = 32.0f / (float)OH;             // 1/16
    float sy = ((float)oy + 0.5f) * scale - 0.5f;
    float sx = ((float)ox + 0.5f) * scale - 0.5f;
    int y0 = (int)floorf(sy), x0 = (int)floorf(sx);
    const float fy = sy - (float)y0, fx = sx - (float)x0;
    int y1 = y0 + 1, x1 = x0 + 1;
    y0 = min(max(y0, 0), 31); y1 = min(max(y1, 0), 31);
    x0 = min(max(x0, 0), 31); x1 = min(max(x1, 0), 31);

    const float* src = ood + b * HW;
    const float v00 = src[y0 * 32 + x0], v01 = src[y0 * 32 + x1];
    const float v10 = src[y1 * 32 + x0], v11 = src[y1 * 32 + x1];
    const float top = v00 + fx * (v01 - v00);
    const float bot = v10 + fx * (v11 - v10);
    out[idx] = top + fy * (bot - top);
}

// ---------------------------------------------------------------------------
extern "C" void kernel_launch(void* const* d_in, const int* in_sizes, int n_in,
                              void* d_out, int out_size, void* d_ws, size_t ws_size,
                              hipStream_t stream) {
    const float* emb = (const float*)d_in[0];   // [4,768,32,32]
    const float* db  = (const float*)d_in[1];   // [20000,768]
    // d_in[2]=k(3), d_in[3]=out_h(512), d_in[4]=out_w(512): fixed by reference
    float* out = (float*)d_out;

    float* ws   = (float*)d_ws;
    float* x2   = ws;                  // 20000
    float* q2   = ws + 20480;          // 4096
    float* oodb = ws + 24576;          // 4096
    float* part = ws + 28672;          // 8*4096*3 = 98304

    knn_x2_kernel<<<(NDB + 3) / 4, 256, 0, stream>>>(db, x2);
    knn_q2_kernel<<<NQ / 256, 256, 0, stream>>>(emb, q2);
    knn_main_kernel<<<dim3(NQ / TM, NCHUNKS), 256, 0, stream>>>(emb, db, q2, x2, part);
    knn_merge_kernel<<<NQ / 256, 256, 0, stream>>>(part, oodb);
    knn_upsample_kernel<<<(BATCH * OH * OW) / 256, 256, 0, stream>>>(oodb, out);
}